// JTNNEncoder_3934190044269
// MI455X (gfx1250) — compile-verified
//
#include <hip/hip_runtime.h>

#define HID    256
#define MAX_NB 8
#define EPD    8192
#define DLEV   24
#define ETOT   (DLEV * EPD)     // 196608
#define NB     8192             // roots == EPD

typedef float v2f __attribute__((ext_vector_type(2)));
typedef float v8f __attribute__((ext_vector_type(8)));

// D = A(16x4 f32) * B(4x16 f32) + C(16x16 f32)
// A frag: lane<16 holds {A[m][k0],A[m][k0+1]}, lane>=16 holds {A[m][k0+2],A[m][k0+3]}
// B frag: lane<16 holds {B[k0][n],B[k0+1][n]},  lane>=16 holds {B[k0+2][n],B[k0+3][n]}
__device__ __forceinline__ v8f wmma4(float a0, float a1, float b0, float b1, v8f c) {
    v2f a = {a0, a1};
    v2f b = {b0, b1};
    return __builtin_amdgcn_wmma_f32_16x16x4_f32(false, a, false, b, (short)0, c, false, false);
}

__device__ __forceinline__ float sigmoidf_(float x) { return 1.0f / (1.0f + __expf(-x)); }

// ---------------------------------------------------------------------------
// Weight transpose: WT[n*K + k] = W[k*256 + n]   (one-time, 2 MB total)
// ---------------------------------------------------------------------------
__global__ __launch_bounds__(256) void transpose_kernel(
    const float* __restrict__ W, float* __restrict__ WT, int K)
{
    const int idx = blockIdx.x * 256 + threadIdx.x;   // grid = K blocks
    const int n = idx / K;
    const int k = idx % K;
    WT[(size_t)n * K + k] = W[(size_t)k * HID + n];
}

// ---------------------------------------------------------------------------
// Gather: X[n] = emb[wid[n]];  sumH[n] = sum_j h[nei[n][j]]  (nei==ETOT -> 0)
// 256 threads = 4 messages x 64 lanes (float4 per lane). Grid = 2048.
// ---------------------------------------------------------------------------
__global__ __launch_bounds__(256) void gather_kernel(
    const int* __restrict__ wid, const int* __restrict__ nei,
    const float* __restrict__ emb, const float* __restrict__ h,
    float* __restrict__ X, float* __restrict__ sumH)
{
    const int tid = threadIdx.x;
    const int msg = blockIdx.x * 4 + (tid >> 6);
    const int l   = tid & 63;

    const int w = wid[msg];
    const float4 xv = *(const float4*)(emb + (size_t)w * HID + l * 4);
    *(float4*)(X + (size_t)msg * HID + l * 4) = xv;

    float s0 = 0.f, s1 = 0.f, s2 = 0.f, s3 = 0.f;
#pragma unroll
    for (int j = 0; j < MAX_NB; ++j) {
        const int idx = nei[msg * MAX_NB + j];
        if (idx != ETOT) {
            const float4 hv = *(const float4*)(h + (size_t)idx * HID + l * 4);
            s0 += hv.x; s1 += hv.y; s2 += hv.z; s3 += hv.w;
        }
    }
    float4 sv; sv.x = s0; sv.y = s1; sv.z = s2; sv.w = s3;
    *(float4*)(sumH + (size_t)msg * HID + l * 4) = sv;
}

// ---------------------------------------------------------------------------
// Generic WMMA GEMM: out = act( [A0 | A1](8192 x K) @ W(K x 256) + bias )
// B matrix supplied TRANSPOSED: WT[n*K + k]  -> float2 (b64) fragment loads
// with compile-time K so all 16 N-tile offsets are immediates.
// MODE 0: none    (XR = X @ Wr)
// MODE 1: sigmoid (Z)
// MODE 2: relu    (roots)
// MODE 3: final   out = (1-z)*sumH + z*tanh(.)   (writes h depth block)
// block = 128 threads (4 waves); wave = 16 rows x 256 cols. Grid = 128.
// ---------------------------------------------------------------------------
template<int K, int MODE>
__global__ __launch_bounds__(128) void gemm_kernel(
    const float* __restrict__ A0, const float* __restrict__ A1,
    const float* __restrict__ WT, const float* __restrict__ bias,
    const float* __restrict__ Zb, const float* __restrict__ sHb,
    float* __restrict__ out)
{
    const int wave  = threadIdx.x >> 5;
    const int lane  = threadIdx.x & 31;
    const int l15   = lane & 15;
    const int khalf = (lane >> 4) << 1;           // 0 or 2
    const int rowBase = blockIdx.x * 64 + wave * 16;
    const int arow    = rowBase + l15;

    v8f acc[16];
#pragma unroll
    for (int nt = 0; nt < 16; ++nt) acc[nt] = (v8f)0.0f;

    const float* wtLane = WT + (size_t)l15 * K;   // row n = l15 (+ nt*16 via imm offset)

    // First 256 K-columns from A0
    {
        const float* arowp = A0 + (size_t)arow * HID;
        for (int k0 = 0; k0 < HID; k0 += 4) {
            const int kk = k0 + khalf;
            const float2 av = *(const float2*)(arowp + kk);
            const float* bp = wtLane + kk;
#pragma unroll
            for (int nt = 0; nt < 16; ++nt) {
                const float2 bv = *(const float2*)(bp + nt * 16 * K); // imm offset
                acc[nt] = wmma4(av.x, av.y, bv.x, bv.y, acc[nt]);
            }
        }
    }
    // Second 256 K-columns from A1 (concat half), only when K == 512
    if (K == 2 * HID) {
        const float* arowp = A1 + (size_t)arow * HID;
        for (int k0 = 0; k0 < HID; k0 += 4) {
            const int kk = k0 + khalf;
            const float2 av = *(const float2*)(arowp + kk);
            const float* bp = wtLane + HID + kk;
#pragma unroll
            for (int nt = 0; nt < 16; ++nt) {
                const float2 bv = *(const float2*)(bp + nt * 16 * K); // imm offset
                acc[nt] = wmma4(av.x, av.y, bv.x, bv.y, acc[nt]);
            }
        }
    }

#pragma unroll
    for (int nt = 0; nt < 16; ++nt) {
        const int col = nt * 16 + l15;
        const float bv = (MODE == 0) ? 0.0f : bias[col];
#pragma unroll
        for (int r = 0; r < 8; ++r) {
            const int grow = rowBase + ((lane < 16) ? r : (r + 8));
            const size_t o = (size_t)grow * HID + col;
            const float v = acc[nt][r] + bv;
            if (MODE == 0) {
                out[o] = v;
            } else if (MODE == 1) {
                out[o] = sigmoidf_(v);
            } else if (MODE == 2) {
                out[o] = fmaxf(v, 0.0f);
            } else { // final GRU combine
                const float z  = Zb[o];
                const float sh = sHb[o];
                out[o] = (1.0f - z) * sh + z * tanhf(v);
            }
        }
    }
}

// ---------------------------------------------------------------------------
// Fused Ur kernel: 16 messages per block (8 waves x 2 msgs):
//   Hn = gather h[nei]  (128 rows x 256, staged in LDS, zero for pad)
//   Rpre = Hn @ Ur      (WMMA; UrT staged in LDS in two 128-K chunks)
//   r = sigmoid(XR[msg] + Rpre + Ur_b); sumG[msg] = sum_j r * Hn
// LDS: A 128x260 + BT 256x132 floats = 262 KB (dynamic). Grid = 512 x 256.
// ---------------------------------------------------------------------------
#define LDS_A_STRIDE  260
#define LDS_BT_STRIDE 132

__global__ __launch_bounds__(256) void ur_gate_kernel(
    const int* __restrict__ nei, const float* __restrict__ h,
    const float* __restrict__ XR, const float* __restrict__ UrT,
    const float* __restrict__ Urb, float* __restrict__ sumG)
{
    extern __shared__ float lds[];
    float* ldsA = lds;                               // 128 x LDS_A_STRIDE (rows x K)
    float* ldsB = lds + 128 * LDS_A_STRIDE;          // 256 x LDS_BT_STRIDE (n x Kchunk)

    const int tid   = threadIdx.x;
    const int wave  = tid >> 5;
    const int lane  = tid & 31;
    const int l15   = lane & 15;
    const int khalf = (lane >> 4) << 1;
    const int m0    = blockIdx.x * 16 + wave * 2;    // this wave: messages m0, m0+1

    // Stage this wave's 16 neighbor rows (rows 0-7 = m0, 8-15 = m0+1)
#pragma unroll
    for (int r = 0; r < 16; ++r) {
        const int msg = m0 + (r >> 3);
        const int idx = nei[msg * MAX_NB + (r & 7)];
        float* dst = ldsA + (wave * 16 + r) * LDS_A_STRIDE;
        if (idx != ETOT) {
            const float* src = h + (size_t)idx * HID;
            for (int v = lane; v < 64; v += 32)
                *(float4*)(dst + v * 4) = *(const float4*)(src + v * 4);
        } else {
            const float4 zz = {0.f, 0.f, 0.f, 0.f};
            for (int v = lane; v < 64; v += 32)
                *(float4*)(dst + v * 4) = zz;
        }
    }

    v8f acc[16];
#pragma unroll
    for (int nt = 0; nt < 16; ++nt) acc[nt] = (v8f)0.0f;

    for (int c = 0; c < 2; ++c) {
        __syncthreads();
        // cooperative load: ldsB[n][kl] = UrT[n*256 + c*128 + kl], 256 x 128
        for (int i = tid; i < 256 * 32; i += 256) {
            const int n = i >> 5, v = i & 31;
            const float4 t = *(const float4*)(UrT + (size_t)n * HID + c * 128 + v * 4);
            *(float4*)(ldsB + n * LDS_BT_STRIDE + v * 4) = t;
        }
        __syncthreads();
        const float* btLane = ldsB + l15 * LDS_BT_STRIDE;
        for (int kl = 0; kl < 128; kl += 4) {
            const int kk = kl + khalf;
            const float* ap = ldsA + (wave * 16 + l15) * LDS_A_STRIDE + (c * 128 + kk);
            const float a0 = ap[0], a1 = ap[1];
            const float* bp = btLane + kk;
#pragma unroll
            for (int nt = 0; nt < 16; ++nt) {
                const float2 bv = *(const float2*)(bp + nt * 16 * LDS_BT_STRIDE);
                acc[nt] = wmma4(a0, a1, bv.x, bv.y, acc[nt]);
            }
        }
    }

    // Epilogue: lanes 0-15 own m0 (tile rows 0-7), lanes 16-31 own m0+1 (rows 8-15)
    const int msg = m0 + (lane >> 4);
#pragma unroll
    for (int nt = 0; nt < 16; ++nt) {
        const int col = nt * 16 + l15;
        const float xr = XR[(size_t)msg * HID + col];
        const float ub = Urb[col];
        float s = 0.0f;
#pragma unroll
        for (int r = 0; r < 8; ++r) {
            const int rowT = (lane < 16) ? r : (r + 8);
            const float hv = ldsA[(wave * 16 + rowT) * LDS_A_STRIDE + col];
            const float g  = sigmoidf_(acc[nt][r] + xr + ub);
            s += g * hv;
        }
        sumG[(size_t)msg * HID + col] = s;
    }
}

// ---------------------------------------------------------------------------
extern "C" void kernel_launch(void* const* d_in, const int* in_sizes, int n_in,
                              void* d_out, int out_size, void* d_ws, size_t ws_size,
                              hipStream_t stream)
{
    (void)in_sizes; (void)n_in; (void)out_size; (void)ws_size;

    const float* emb  = (const float*)d_in[0];
    const float* Wz_w = (const float*)d_in[1];
    const float* Wz_b = (const float*)d_in[2];
    const float* Wr_w = (const float*)d_in[3];
    const float* Ur_w = (const float*)d_in[4];
    const float* Ur_b = (const float*)d_in[5];
    const float* Wh_w = (const float*)d_in[6];
    const float* Wh_b = (const float*)d_in[7];
    const float* W_w  = (const float*)d_in[8];
    const float* W_b  = (const float*)d_in[9];
    const int* msg_wid  = (const int*)d_in[10];
    const int* msg_nei  = (const int*)d_in[11];
    const int* root_wid = (const int*)d_in[12];
    const int* root_nei = (const int*)d_in[13];

    float* h     = (float*)d_out;                     // h[:E] lives in d_out
    float* roots = h + (size_t)ETOT * HID;            // root_vecs section

    float* X    = (float*)d_ws;
    float* sumH = X    + (size_t)NB * HID;
    float* Z    = sumH + (size_t)NB * HID;
    float* XR   = Z    + (size_t)NB * HID;
    float* sumG = XR   + (size_t)NB * HID;            // 5 x 8 MB
    float* WzT  = sumG + (size_t)NB * HID;            // 256 x 512
    float* WrT  = WzT  + (size_t)512 * HID;           // 256 x 256
    float* UrT  = WrT  + (size_t)256 * HID;           // 256 x 256
    float* WhT  = UrT  + (size_t)256 * HID;           // 256 x 512
    float* WoT  = WhT  + (size_t)512 * HID;           // 256 x 512   (~42 MB total)

    const size_t ldsBytes =
        (size_t)(128 * LDS_A_STRIDE + 256 * LDS_BT_STRIDE) * sizeof(float);

    // One-time weight transposes (cheap: 2 MB of traffic)
    transpose_kernel<<<512, 256, 0, stream>>>(Wz_w, WzT, 2 * HID);
    transpose_kernel<<<256, 256, 0, stream>>>(Wr_w, WrT, HID);
    transpose_kernel<<<256, 256, 0, stream>>>(Ur_w, UrT, HID);
    transpose_kernel<<<512, 256, 0, stream>>>(Wh_w, WhT, 2 * HID);
    transpose_kernel<<<512, 256, 0, stream>>>(W_w,  WoT, 2 * HID);

    for (int t = 0; t < DLEV; ++t) {
        const int* wid = msg_wid + (size_t)t * EPD;
        const int* nei = msg_nei + (size_t)t * EPD * MAX_NB;

        gather_kernel<<<2048, 256, 0, stream>>>(wid, nei, emb, h, X, sumH);
        gemm_kernel<512, 1><<<128, 128, 0, stream>>>(X, sumH, WzT, Wz_b, nullptr, nullptr, Z);
        gemm_kernel<256, 0><<<128, 128, 0, stream>>>(X, X,    WrT, nullptr, nullptr, nullptr, XR);
        ur_gate_kernel<<<512, 256, ldsBytes, stream>>>(nei, h, XR, UrT, Ur_b, sumG);
        gemm_kernel<512, 3><<<128, 128, 0, stream>>>(X, sumG, WhT, Wh_b, Z, sumH,
                                                     h + (size_t)t * EPD * HID);
    }

    // Root aggregation: relu([emb[root_wid] | sum h[root_nei]] @ W + b)
    gather_kernel<<<2048, 256, 0, stream>>>(root_wid, root_nei, emb, h, X, sumH);
    gemm_kernel<512, 2><<<128, 128, 0, stream>>>(X, sumH, WoT, W_b, nullptr, nullptr, roots);
}